// SGP_LAYER_77807627534630
// MI455X (gfx1250) — compile-verified
//
#include <hip/hip_runtime.h>

// ---------------------------------------------------------------------------
// Sparse-GP layer on MI455X (gfx1250, wave32).
// All GEMMs: V_WMMA_F32_16X16X4_F32 with 4x2 register blocking
// (one wave = 64x32 f32 macro-tile, 8 accumulators = 64 VGPRs -> no spills,
// 8 WMMAs per 6 fragment loads). Triangular solves replaced by explicit
// 128x128 L^-1; 64KB global->LDS staging uses GLOBAL_LOAD_ASYNC_TO_LDS_B128.
// ---------------------------------------------------------------------------

typedef float v8f __attribute__((ext_vector_type(8)));
typedef float v2f __attribute__((ext_vector_type(2)));
typedef int   i32x4 __attribute__((vector_size(4 * sizeof(int))));

#define B_    32
#define L_    1024
#define H_    12
#define DV_   64
#define MH_   64
#define M_    128
#define HDIM_ 768
#define QVD_  1536
#define BH_   (B_ * H_)
#define JITTER 1e-6f
#define INV_SQRT_M 0.08838834764831845f   // 1/sqrt(128)

// gfx1250 async global->LDS path (ASYNCcnt); guarded fallback to plain stores.
#if __has_builtin(__builtin_amdgcn_global_load_async_to_lds_b128) && \
    __has_builtin(__builtin_amdgcn_s_wait_asynccnt)
#define HAVE_ASYNC_LDS 1
typedef __attribute__((address_space(1))) i32x4* gp_i32x4;
typedef __attribute__((address_space(3))) i32x4* sp_i32x4;
#define AS1P(p) ((gp_i32x4)(unsigned long long)(p))
#define AS3P(p) ((sp_i32x4)(unsigned int)(unsigned long long)(p))
#else
#define HAVE_ASYNC_LDS 0
#endif

// ---- WMMA 4x2 block --------------------------------------------------------
// acc(64x32) += A(64xK) * B(Kx32); A(m,k) at A[m*am+k*ak], B(k,n) at B[k*bk+n*bn].
// fp32 fragment layout (ISA 7.12.2): A: lane=(m%16), vgpr r -> k = r + 2*(lane/16)
//                                    B: lane=(n%16), vgpr r -> k = r + 2*(lane/16)
__device__ __forceinline__ void wmma_block4x2(const float* __restrict__ A, long am, long ak,
                                              const float* __restrict__ B, long bk, long bn,
                                              int K, v8f acc[4][2]) {
  const int lane = threadIdx.x & 31;
  const int half = lane >> 4;
  const int l16  = lane & 15;
  const float* ap = A + (long)l16 * am + (long)(2 * half) * ak;
  const float* bp = B + (long)l16 * bn + (long)(2 * half) * bk;
  for (int k = 0; k < K; k += 4) {
    v2f a[4], b[2];
#pragma unroll
    for (int t = 0; t < 4; ++t) {
      const float* at = ap + (long)(t * 16) * am;
      a[t].x = at[0]; a[t].y = at[ak];
    }
#pragma unroll
    for (int t = 0; t < 2; ++t) {
      const float* bt = bp + (long)(t * 16) * bn;
      b[t].x = bt[0]; b[t].y = bt[bk];
    }
#pragma unroll
    for (int i = 0; i < 4; ++i)
#pragma unroll
      for (int j = 0; j < 2; ++j)
        acc[i][j] = __builtin_amdgcn_wmma_f32_16x16x4_f32(false, a[i], false, b[j],
                                                          (short)0, acc[i][j], false, false);
    ap += 4 * ak;
    bp += 4 * bk;
  }
}

// C/D layout: acc[e] for lane l -> row = e + 8*(l/16), col = l%16
__device__ __forceinline__ void store_block(float* __restrict__ C, long ldc, v8f acc[4][2]) {
  const int lane = threadIdx.x & 31;
  const int half = lane >> 4;
  const int l16  = lane & 15;
#pragma unroll
  for (int i = 0; i < 4; ++i)
#pragma unroll
    for (int j = 0; j < 2; ++j)
#pragma unroll
      for (int e = 0; e < 8; ++e)
        C[(long)(i * 16 + e + 8 * half) * ldc + j * 16 + l16] = acc[i][j][e];
}

// ---- stage 0: omega scaling + per-head amplitude ---------------------------
__global__ void k_prep(const float* __restrict__ omega_half,
                       const float* __restrict__ log_ls,
                       const float* __restrict__ log_sf,
                       float* __restrict__ omega_s,
                       float* __restrict__ amp) {
  int idx = blockIdx.x * blockDim.x + threadIdx.x;
  if (idx < H_ * DV_ * MH_) {
    int h = idx / (DV_ * MH_);
    int d = (idx / MH_) % DV_;
    omega_s[idx] = omega_half[idx] * (1.41421356237f * __expf(-log_ls[h * DV_ + d]));
  }
  if (idx < H_) amp[idx] = sqrtf(__expf(log_sf[idx]));
}

// ---- stage 1: qv = x @ W_qv^T   (32768 x 1536, K=768) ----------------------
__global__ void __launch_bounds__(256, 1)
k_gemm_qv(const float* __restrict__ x,
          const float* __restrict__ Wqv,
          float* __restrict__ qv) {
  const int NT = QVD_ / 32;                       // 48 macro-cols
  int wave = blockIdx.x * 8 + (threadIdx.x >> 5); // 24576 waves total
  int tm = wave / NT, tn = wave % NT;
  long m0 = (long)tm * 64, n0 = (long)tn * 32;
  v8f acc[4][2] = {};
  // B(k,n) = Wqv[n*768 + k]
  wmma_block4x2(x + m0 * HDIM_, HDIM_, 1, Wqv + n0 * HDIM_, 1, HDIM_, HDIM_, acc);
  store_block(qv + m0 * QVD_ + n0, QVD_, acc);
}

// ---- stage 2: proj = q @ omega_s + phase ; phi = amp*[cos,sin]/sqrt(M) -----
__global__ void __launch_bounds__(256, 1)
k_proj_phi(const float* __restrict__ qv,
           const float* __restrict__ omega_s,
           const float* __restrict__ phase,
           const float* __restrict__ amp,
           float* __restrict__ phi) {
  int bh = blockIdx.y;
  int b = bh / H_, h = bh % H_;
  int wave = blockIdx.x * 8 + (threadIdx.x >> 5);   // 32 waves per bh: 16x2 macros
  int m0 = (wave >> 1) * 64, n0 = (wave & 1) * 32;
  const float* q  = qv + (long)b * L_ * QVD_ + (long)h * (2 * DV_);  // q(l,d): ld=QVD_
  const float* om = omega_s + (long)h * DV_ * MH_;                    // om(d,m): ld=MH_
  v8f acc[4][2] = {};
  wmma_block4x2(q + (long)m0 * QVD_, QVD_, 1, om + n0, MH_, 1, DV_, acc);

  const int lane = threadIdx.x & 31;
  const int half = lane >> 4;
  const int l16  = lane & 15;
  const float a  = amp[h] * INV_SQRT_M;
  float ph[2];
#pragma unroll
  for (int j = 0; j < 2; ++j) ph[j] = phase[h * MH_ + n0 + j * 16 + l16];
  float* pout = phi + (long)bh * L_ * M_;
#pragma unroll
  for (int i = 0; i < 4; ++i)
#pragma unroll
    for (int j = 0; j < 2; ++j)
#pragma unroll
      for (int e = 0; e < 8; ++e) {
        long row = m0 + i * 16 + e + 8 * half;
        int  cn  = n0 + j * 16 + l16;                  // feature in [0,64)
        float p = acc[i][j][e] + ph[j];
        float s, c;
        __sincosf(p, &s, &c);
        pout[row * M_ + cn]       = a * c;   // cos half [0,64)
        pout[row * M_ + cn + MH_] = a * s;   // sin half [64,128)
      }
}

// ---- stage 3: A = phi^T phi + (sigma2+jitter) I  (128x128, K=1024) ---------
__global__ void __launch_bounds__(256, 1)
k_gram(const float* __restrict__ phi,
       const float* __restrict__ log_sigma2,
       float* __restrict__ Amat) {
  int bh = blockIdx.y;
  int wave = threadIdx.x >> 5;            // 8 waves: 2x4 macro grid
  int m0 = (wave >> 2) * 64, n0 = (wave & 3) * 32;
  const float* p = phi + (long)bh * L_ * M_;
  v8f acc[4][2] = {};
  // G(m,n) = sum_l p[l*128+m] * p[l*128+n]
  wmma_block4x2(p + m0, 1, M_, p + n0, M_, 1, L_, acc);
  const float lam = __expf(log_sigma2[0]) + JITTER;
  const int lane = threadIdx.x & 31;
  const int half = lane >> 4;
  const int l16  = lane & 15;
  float* c = Amat + (long)bh * M_ * M_;
#pragma unroll
  for (int i = 0; i < 4; ++i)
#pragma unroll
    for (int j = 0; j < 2; ++j)
#pragma unroll
      for (int e = 0; e < 8; ++e) {
        int gm = m0 + i * 16 + e + 8 * half, gn = n0 + j * 16 + l16;
        float v = acc[i][j][e];
        if (gm == gn) v += lam;
        c[(long)gm * M_ + gn] = v;
      }
}

// ---- stage 4: in-LDS Cholesky of A (128x128) per (b,h); logdet -------------
__global__ void k_chol(float* __restrict__ Amat, float* __restrict__ logdet) {
  __shared__ float Ls[M_][M_];          // 64 KB
  const int bh = blockIdx.x;
  const int i  = threadIdx.x;           // row owner, blockDim = 128
  float* Ag = Amat + (long)bh * M_ * M_;
#if HAVE_ASYNC_LDS
  for (int it = 0; it < 32; ++it) {     // 128 lanes x 16B x 32 = 64 KB
    long fo = ((long)it * 128 + i) * 4;
    __builtin_amdgcn_global_load_async_to_lds_b128(AS1P(Ag + fo), AS3P(&Ls[0][0] + fo), 0, 0);
  }
  __builtin_amdgcn_s_wait_asynccnt(0);
#else
  for (int j = 0; j < M_; ++j) Ls[i][j] = Ag[(long)i * M_ + j];
#endif
  __syncthreads();
  for (int k = 0; k < M_; ++k) {
    float diag = sqrtf(Ls[k][k]);       // all read before any write (sync below)
    __syncthreads();
    if (i >= k) Ls[i][k] /= diag;       // thread k sets Ls[k][k] = sqrt(akk)
    __syncthreads();
    if (i > k) {
      float lik = Ls[i][k];
      for (int j = k + 1; j <= i; ++j) Ls[i][j] -= lik * Ls[j][k];
    }
    __syncthreads();
  }
  for (int j = 0; j < M_; ++j) Ag[(long)i * M_ + j] = (j <= i) ? Ls[i][j] : 0.0f;
  float ld = __logf(Ls[i][i]);
  __syncthreads();
  float* red = &Ls[0][0];
  red[i] = ld;
  __syncthreads();
  for (int s = 64; s > 0; s >>= 1) {
    if (i < s) red[i] += red[i + s];
    __syncthreads();
  }
  if (i == 0) logdet[bh] = 2.0f * red[0];
}

// ---- stage 5: Linv = L^-1 (forward substitution, one column per thread) ----
__global__ void k_trinv(const float* __restrict__ Lmat,
                        float* __restrict__ Linv,
                        float* __restrict__ trAinv) {
  __shared__ float Ls[M_][M_];          // 64 KB
  const int bh = blockIdx.x;
  const int j  = threadIdx.x;           // column owner, blockDim = 128
  const float* Lg = Lmat + (long)bh * M_ * M_;
#if HAVE_ASYNC_LDS
  for (int it = 0; it < 32; ++it) {
    long fo = ((long)it * 128 + j) * 4;
    __builtin_amdgcn_global_load_async_to_lds_b128(AS1P(Lg + fo), AS3P(&Ls[0][0] + fo), 0, 0);
  }
  __builtin_amdgcn_s_wait_asynccnt(0);
#else
  for (int r = 0; r < M_; ++r) Ls[r][j] = Lg[(long)r * M_ + j];
#endif
  __syncthreads();
  float xcol[M_];                        // column j of Linv (scratch-backed)
  float* Xg = Linv + (long)bh * M_ * M_;
  float ss = 0.0f;
  for (int r = 0; r < j; ++r) Xg[(long)r * M_ + j] = 0.0f;
  for (int r = j; r < M_; ++r) {
    float s = (r == j) ? 1.0f : 0.0f;
    for (int t = j; t < r; ++t) s -= Ls[r][t] * xcol[t];
    float v = s / Ls[r][r];
    xcol[r] = v;
    Xg[(long)r * M_ + j] = v;
    ss += v * v;
  }
  __syncthreads();
  float* red = &Ls[0][0];
  red[j] = ss;
  __syncthreads();
  for (int s = 64; s > 0; s >>= 1) {
    if (j < s) red[j] += red[j + s];
    __syncthreads();
  }
  if (j == 0) trAinv[bh] = red[0];       // sum(Linv^2) == sum(Rinv^2)
}

// ---- stage 6: tQ = phi @ Linv^T (1024x128, K=128); fused rowsum(tQ^2) ------
__global__ void __launch_bounds__(256, 1)
k_tq(const float* __restrict__ phi,
     const float* __restrict__ Linv,
     float* __restrict__ tQ,
     float* __restrict__ varacc) {
  int bh = blockIdx.y;
  int wave = blockIdx.x * 8 + (threadIdx.x >> 5);   // 64 waves per bh: 16x4 macros
  int m0 = (wave >> 2) * 64, n0 = (wave & 3) * 32;
  const float* p = phi  + (long)bh * L_ * M_;
  const float* X = Linv + (long)bh * M_ * M_;
  v8f acc[4][2] = {};
  // tQ(l,m) = sum_t phi(l,t) * Linv(m,t): B(k,n) = X[(n0+n)*128 + k]
  wmma_block4x2(p + (long)m0 * M_, M_, 1, X + (long)n0 * M_, 1, M_, M_, acc);
  store_block(tQ + (long)bh * L_ * M_ + (long)m0 * M_ + n0, M_, acc);

  // partial row sum of squares over this wave's 32 columns
  const int lane = threadIdx.x & 31;
  const int half = lane >> 4;
  const int l16  = lane & 15;
#pragma unroll
  for (int i = 0; i < 4; ++i)
#pragma unroll
    for (int e = 0; e < 8; ++e) {
      float rs = 0.0f;
#pragma unroll
      for (int j = 0; j < 2; ++j) rs += acc[i][j][e] * acc[i][j][e];
#pragma unroll
      for (int s = 1; s < 16; s <<= 1) rs += __shfl_xor(rs, s, 32);  // within 16-lane half
      if (l16 == 0)
        atomicAdd(&varacc[(long)bh * L_ + m0 + i * 16 + e + 8 * half], rs);
    }
}

// ---- stage 7: S = tQ^T v  (128x64, K=1024) ---------------------------------
__global__ void __launch_bounds__(128, 1)
k_s(const float* __restrict__ tQ,
    const float* __restrict__ qv,
    float* __restrict__ Sm) {
  int bh = blockIdx.y;
  int b = bh / H_, h = bh % H_;
  int wave = threadIdx.x >> 5;          // 4 waves per bh: 2x2 macros
  int m0 = (wave >> 1) * 64, n0 = (wave & 1) * 32;
  const float* tq = tQ + (long)bh * L_ * M_;
  const float* v  = qv + (long)b * L_ * QVD_ + (long)h * (2 * DV_) + DV_;  // v(l,d): ld=QVD_
  v8f acc[4][2] = {};
  // S(m,d) = sum_l tq[l*128+m] * v[l*QVD_+d]
  wmma_block4x2(tq + m0, 1, M_, v + n0, QVD_, 1, L_, acc);
  store_block(Sm + (long)bh * M_ * DV_ + (long)m0 * DV_ + n0, DV_, acc);
}

// ---- stage 7b: std = sqrt(max(0, sigma2 * varacc)) in place ----------------
__global__ void k_zero_var(float* __restrict__ varacc) {
  long idx = (long)blockIdx.x * blockDim.x + threadIdx.x;
  if (idx < (long)BH_ * L_) varacc[idx] = 0.0f;
}
__global__ void k_std(float* __restrict__ stdv, const float* __restrict__ log_sigma2) {
  long idx = (long)blockIdx.x * blockDim.x + threadIdx.x;
  if (idx >= (long)BH_ * L_) return;
  float sigma2 = __expf(log_sigma2[0]);
  stdv[idx] = sqrtf(fmaxf(sigma2 * stdv[idx], 0.0f));
}

// ---- stage 8: mean = tQ @ S ; samp = mean + std*eps (H-interleaved) --------
__global__ void __launch_bounds__(256, 1)
k_mean_samp(const float* __restrict__ tQ,
            const float* __restrict__ Sm,
            const float* __restrict__ stdv,
            const float* __restrict__ eps,
            float* __restrict__ samp) {
  int bh = blockIdx.y;
  int b = bh / H_, h = bh % H_;
  int wave = blockIdx.x * 8 + (threadIdx.x >> 5);   // 32 waves per bh: 16x2 macros
  int m0 = (wave >> 1) * 64, n0 = (wave & 1) * 32;
  const float* tq = tQ + (long)bh * L_ * M_;
  const float* S  = Sm + (long)bh * M_ * DV_;
  v8f acc[4][2] = {};
  wmma_block4x2(tq + (long)m0 * M_, M_, 1, S + n0, DV_, 1, M_, acc);
  const int lane = threadIdx.x & 31;
  const int half = lane >> 4;
  const int l16  = lane & 15;
#pragma unroll
  for (int i = 0; i < 4; ++i)
#pragma unroll
    for (int j = 0; j < 2; ++j)
#pragma unroll
      for (int e = 0; e < 8; ++e) {
        long row = m0 + i * 16 + e + 8 * half;          // l
        int  d   = n0 + j * 16 + l16;
        float sd = stdv[(long)bh * L_ + row];
        float ev = eps[((long)bh * L_ + row) * DV_ + d];
        samp[((long)b * L_ + row) * HDIM_ + (long)h * DV_ + d] = acc[i][j][e] + sd * ev;
      }
}

// ---- stage 9: out = samp @ W_O^T + b  (32768 x 768, K=768) -----------------
__global__ void __launch_bounds__(256, 1)
k_out(const float* __restrict__ samp,
      const float* __restrict__ Wo,
      const float* __restrict__ bo,
      float* __restrict__ out) {
  const int NT = HDIM_ / 32;                        // 24 macro-cols
  int wave = blockIdx.x * 8 + (threadIdx.x >> 5);   // 12288 waves total
  int tm = wave / NT, tn = wave % NT;
  long m0 = (long)tm * 64, n0 = (long)tn * 32;
  v8f acc[4][2] = {};
  wmma_block4x2(samp + m0 * HDIM_, HDIM_, 1, Wo + n0 * HDIM_, 1, HDIM_, HDIM_, acc);
  const int lane = threadIdx.x & 31;
  const int half = lane >> 4;
  const int l16  = lane & 15;
  float bias[2];
#pragma unroll
  for (int j = 0; j < 2; ++j) bias[j] = bo[n0 + j * 16 + l16];
#pragma unroll
  for (int i = 0; i < 4; ++i)
#pragma unroll
    for (int j = 0; j < 2; ++j)
#pragma unroll
      for (int e = 0; e < 8; ++e)
        out[(m0 + i * 16 + e + 8 * half) * HDIM_ + n0 + j * 16 + l16] = acc[i][j][e] + bias[j];
}

// ---- stage 10: KL ----------------------------------------------------------
__global__ void k_zero(float* p) { if (threadIdx.x == 0 && blockIdx.x == 0) *p = 0.0f; }

__global__ void k_kl(const float* __restrict__ Linv,
                     const float* __restrict__ Sm,
                     const float* __restrict__ trAinv,
                     const float* __restrict__ logdet,
                     const float* __restrict__ log_sigma2,
                     float* __restrict__ kl_out) {
  __shared__ float Ssh[M_ * DV_];       // 32 KB
  __shared__ float red[256];
  const int bh  = blockIdx.x;
  const int tid = threadIdx.x;          // blockDim = 256
  const float* Sg = Sm + (long)bh * M_ * DV_;
  const float* Lg = Linv + (long)bh * M_ * M_;
  for (int t = tid; t < M_ * DV_; t += 256) Ssh[t] = Sg[t];
  __syncthreads();
  // W*(m,d) = sum_t Linv(t,m) * S(t,d); thread owns m = tid%128, d in 32-chunk
  const int m  = tid & (M_ - 1);
  const int d0 = (tid >> 7) * 32;
  float ws[32];
  for (int i = 0; i < 32; ++i) ws[i] = 0.0f;
  for (int t = 0; t < M_; ++t) {
    float lv = Lg[(long)t * M_ + m];
    const float* srow = &Ssh[t * DV_ + d0];
    for (int i = 0; i < 32; ++i) ws[i] += lv * srow[i];
  }
  float ss = 0.0f;
  for (int i = 0; i < 32; ++i) ss += ws[i] * ws[i];
  red[tid] = ss;
  __syncthreads();
  for (int s = 128; s > 0; s >>= 1) {
    if (tid < s) red[tid] += red[tid + s];
    __syncthreads();
  }
  if (tid == 0) {
    float sigma2 = __expf(log_sigma2[0]);
    float kl = 0.5f * ((float)DV_ * sigma2 * trAinv[bh] + red[0]
                       - (float)(M_ * DV_)
                       - (float)DV_ * ((float)M_ * __logf(sigma2) - logdet[bh]));
    atomicAdd(kl_out, kl * (1.0f / (float)B_));
  }
}

// ---------------------------------------------------------------------------
extern "C" void kernel_launch(void* const* d_in, const int* in_sizes, int n_in,
                              void* d_out, int out_size, void* d_ws, size_t ws_size,
                              hipStream_t stream) {
  const float* x          = (const float*)d_in[0];
  const float* W_qv       = (const float*)d_in[1];
  const float* log_sf     = (const float*)d_in[2];
  const float* log_ls     = (const float*)d_in[3];
  const float* log_sigma2 = (const float*)d_in[4];
  const float* omega_half = (const float*)d_in[5];
  const float* phase      = (const float*)d_in[6];
  const float* W_O_w      = (const float*)d_in[7];
  const float* W_O_b      = (const float*)d_in[8];
  const float* eps        = (const float*)d_in[9];
  float* out = (float*)d_out;

  // workspace layout (floats)
  float* w = (float*)d_ws;
  size_t off = 0;
  float* qv      = w + off; off += (size_t)B_ * L_ * QVD_;   // 50331648
  float* omega_s = w + off; off += (size_t)H_ * DV_ * MH_;   // 49152
  float* ampv    = w + off; off += 16;
  float* phi     = w + off; off += (size_t)BH_ * L_ * M_;    // 50331648
  float* Amat    = w + off; off += (size_t)BH_ * M_ * M_;    // 6291456 (A, then L in-place)
  float* Linv    = w + off; off += (size_t)BH_ * M_ * M_;    // 6291456
  float* tQ      = w + off; off += (size_t)BH_ * L_ * M_;    // 50331648
  float* Sm      = w + off; off += (size_t)BH_ * M_ * DV_;   // 3145728
  float* stdv    = w + off; off += (size_t)BH_ * L_;         // 393216 (varacc -> std)
  float* logdet  = w + off; off += BH_;
  float* trainv  = w + off; off += BH_;
  float* samp    = qv;  // qv (v half) dead after k_s; safe alias for k_mean_samp

  k_prep<<<(H_ * DV_ * MH_ + 255) / 256, 256, 0, stream>>>(omega_half, log_ls, log_sf,
                                                           omega_s, ampv);
  // qv GEMM: 512 x 48 macro-tiles, 8 waves/block
  k_gemm_qv<<<(512 * 48) / 8, 256, 0, stream>>>(x, W_qv, qv);
  // proj+phi: 16x2 macros per (b,h)
  k_proj_phi<<<dim3(4, BH_), 256, 0, stream>>>(qv, omega_s, phase, ampv, phi);
  // Gram: 2x4 macros per (b,h), 8 waves
  k_gram<<<dim3(1, BH_), 256, 0, stream>>>(phi, log_sigma2, Amat);
  k_chol<<<BH_, 128, 0, stream>>>(Amat, logdet);
  k_trinv<<<BH_, 128, 0, stream>>>(Amat, Linv, trainv);
  // tQ (+ fused row variance accumulation): 16x4 macros per (b,h)
  k_zero_var<<<((size_t)BH_ * L_ + 255) / 256, 256, 0, stream>>>(stdv);
  k_tq<<<dim3(8, BH_), 256, 0, stream>>>(phi, Linv, tQ, stdv);
  // S: 2x2 macros per (b,h)
  k_s<<<dim3(1, BH_), 128, 0, stream>>>(tQ, qv, Sm);
  k_std<<<((size_t)BH_ * L_ + 255) / 256, 256, 0, stream>>>(stdv, log_sigma2);
  // mean + sample: 16x2 macros per (b,h)
  k_mean_samp<<<dim3(4, BH_), 256, 0, stream>>>(tQ, Sm, stdv, eps, samp);
  // output GEMM: 512 x 24 macro-tiles
  k_out<<<(512 * 24) / 8, 256, 0, stream>>>(samp, W_O_w, W_O_b, out);
  // KL scalar at out[B*L*HDIM]
  float* kl_out = out + (size_t)B_ * L_ * HDIM_;
  k_zero<<<1, 32, 0, stream>>>(kl_out);
  k_kl<<<BH_, 256, 0, stream>>>(Linv, Sm, trainv, logdet, log_sigma2, kl_out);
}